// AnisoDiff2D_45973329937291
// MI455X (gfx1250) — compile-verified
//
#include <hip/hip_runtime.h>
#include <cstdint>

// ---------------------------------------------------------------------------
// Perona-Malik anisotropic diffusion (option 2), 10 sweeps, fp32.
// Strategy: one kernel launch per sweep, ping-pong d_ws <-> d_out (both 48MiB;
// the pair fits in the MI455X's 192MB L2, so sweeps 2..10 are L2-resident).
// Each block stages a 66x18 halo tile into LDS via the CDNA5 async
// global->LDS copy path (global_load_async_to_lds_b128/_b32, ASYNCcnt,
// s_wait_asynccnt), then computes 64x16 outputs from LDS with a rolling
// 3x3 register window and v_rcp_f32 for the nonlinear coefficient.
// WMMA is intentionally NOT used: the per-tap coefficient 1/(1+(nab/k)^2)
// makes the stencil nonlinear, so it cannot be expressed as a matmul.
// ---------------------------------------------------------------------------

namespace {
constexpr int BB = 16, CC = 3, HH = 512, WW = 512;
constexpr int NUM_ITER  = 10;
constexpr float DT       = 1.0f / 7.0f;
constexpr float INV_KAPPA = 1.0f / 30.0f;

constexpr int TILE_X = 64;           // outputs per block in x
constexpr int TILE_Y = 16;           // outputs per block in y
constexpr int LDS_S  = 72;           // LDS row stride in floats (288 B, 16B-aligned interior)
constexpr int LDS_H  = TILE_Y + 2;   // 18 rows (halo included)
constexpr int XOFF   = 4;            // interior column 0 at LDS x=4 (byte offset 16)
// loaded columns: image x in [tileX0-1, tileX0+64] -> LDS x in [3, 68]
}

__device__ __forceinline__ void async_ld_b32(uint32_t lds_addr, const float* g) {
  asm volatile("global_load_async_to_lds_b32 %0, %1, off"
               :: "v"(lds_addr), "v"((uint64_t)(uintptr_t)g) : "memory");
}
__device__ __forceinline__ void async_ld_b128(uint32_t lds_addr, const float* g) {
  asm volatile("global_load_async_to_lds_b128 %0, %1, off"
               :: "v"(lds_addr), "v"((uint64_t)(uintptr_t)g) : "memory");
}
__device__ __forceinline__ void wait_async0() {
#if __has_builtin(__builtin_amdgcn_s_wait_asynccnt)
  __builtin_amdgcn_s_wait_asynccnt(0);
#else
  asm volatile("s_wait_asynccnt 0x0" ::: "memory");
#endif
}

__global__ __launch_bounds__(256)
void AnisoDiff2D_pm_step(const float* __restrict__ src,
                         float* __restrict__ dst,
                         const float* __restrict__ factors,   // (8, C)
                         const float* __restrict__ biases)    // (8, C)
{
  __shared__ __align__(16) float tile[LDS_H * LDS_S];

  const int tid    = threadIdx.x;
  const int tileX0 = blockIdx.x * TILE_X;
  const int tileY0 = blockIdx.y * TILE_Y;
  const int plane  = blockIdx.z;              // b*C + c
  const int ch     = plane % CC;
  const float* sp  = src + (size_t)plane * (HH * WW);
  float*       dp  = dst + (size_t)plane * (HH * WW);

  // Per-channel direction constants (block-uniform; 16 cached scalar loads).
  float fk[8], bk[8];
#pragma unroll
  for (int k = 0; k < 8; ++k) {
    fk[k] = factors[k * CC + ch];
    bk[k] = biases [k * CC + ch];
  }

  // ---- 1) zero-fill the LDS tile (implements the zero padding) ------------
  {
    float4* t4 = (float4*)tile;
    constexpr int N4 = (LDS_H * LDS_S) / 4;   // 324
    for (int i = tid; i < N4; i += 256) t4[i] = make_float4(0.f, 0.f, 0.f, 0.f);
  }
  __syncthreads();   // zeros committed before any async LDS write can land

  // ---- 2) async-copy the in-bounds halo tile into LDS ---------------------
  // Interior 64-float row segments: 16 x b128 per row, 18 rows = 288 ops.
  {
    constexpr int NSEG = LDS_H * 16;
    for (int i = tid; i < NSEG; i += 256) {
      const int ly  = i >> 4;          // 0..17
      const int seg = i & 15;          // 0..15
      const int gy  = tileY0 - 1 + ly;
      if ((unsigned)gy < (unsigned)HH) {
        const uint32_t la = (uint32_t)(uintptr_t)&tile[ly * LDS_S + XOFF + seg * 4];
        async_ld_b128(la, sp + (size_t)gy * WW + tileX0 + seg * 4);
      }
    }
  }
  // Left/right halo columns: 2 x b32 per row = 36 ops.
  {
    constexpr int NEDGE = LDS_H * 2;
    for (int i = tid; i < NEDGE; i += 256) {
      const int ly   = i >> 1;
      const int side = i & 1;
      const int gy   = tileY0 - 1 + ly;
      const int gx   = side ? (tileX0 + TILE_X) : (tileX0 - 1);
      const int lx   = side ? (XOFF + TILE_X)   : (XOFF - 1);
      if ((unsigned)gy < (unsigned)HH && (unsigned)gx < (unsigned)WW) {
        const uint32_t la = (uint32_t)(uintptr_t)&tile[ly * LDS_S + lx];
        async_ld_b32(la, sp + (size_t)gy * WW + gx);
      }
    }
  }
  wait_async0();       // this wave's async LDS writes are visible
  __syncthreads();     // ...and published across all 8 waves

  // ---- 3) compute: 4 consecutive rows per thread, rolling 3x3 window ------
  const int tx  = tid & (TILE_X - 1);  // 0..63
  const int tyq = tid >> 6;            // 0..3
  const int r0  = tyq * 4;             // output rows r0..r0+3 (LDS y = r+1)

  constexpr float DIRW[8] = {1.f, 1.f, 1.f, 1.f, 0.5f, 0.5f, 0.5f, 0.5f};

  const int base = XOFF + tx;
  float wN = tile[(r0 + 0) * LDS_S + base - 1];
  float cN = tile[(r0 + 0) * LDS_S + base    ];
  float eN = tile[(r0 + 0) * LDS_S + base + 1];
  float wC = tile[(r0 + 1) * LDS_S + base - 1];
  float cC = tile[(r0 + 1) * LDS_S + base    ];
  float eC = tile[(r0 + 1) * LDS_S + base + 1];

#pragma unroll
  for (int j = 0; j < 4; ++j) {
    const int ls = (r0 + 2 + j) * LDS_S + base;
    const float wS = tile[ls - 1];
    const float cS = tile[ls    ];
    const float eS = tile[ls + 1];

    const float u = cC;
    // direction order: N, S, W, E, NE, SE, SW, NW (matches reference)
    const float nb[8] = { cN, cS, wC, eC, eN, eS, wS, wN };

    float upd = 0.0f;
#pragma unroll
    for (int k = 0; k < 8; ++k) {
      const float nab = __builtin_fmaf(fk[k], nb[k] - u, bk[k]);
      const float t   = nab * INV_KAPPA;
      const float cc  = __builtin_amdgcn_rcpf(__builtin_fmaf(t, t, 1.0f));
      upd = __builtin_fmaf(DIRW[k] * cc, nab, upd);
    }

    dp[(size_t)(tileY0 + r0 + j) * WW + (tileX0 + tx)] = __builtin_fmaf(DT, upd, u);

    // roll the window south
    wN = wC; cN = cC; eN = eC;
    wC = wS; cC = cS; eC = eS;
  }
}

// ---------------------------------------------------------------------------
// Inputs (setup_inputs order): d_in[0]=x (16,3,512,512) f32,
//                              d_in[1]=biases (8,3) f32,
//                              d_in[2]=factors (8,3) f32.
// d_out: (16,3,512,512) f32. d_ws: scratch, needs >= 48 MiB (== out bytes).
// 10 sweeps ping-pong: x -> ws -> out -> ws -> ... -> out (iter 10 in d_out).
// ---------------------------------------------------------------------------
extern "C" void kernel_launch(void* const* d_in, const int* in_sizes, int n_in,
                              void* d_out, int out_size, void* d_ws, size_t ws_size,
                              hipStream_t stream) {
  (void)in_sizes; (void)n_in; (void)out_size; (void)ws_size;

  const float* x       = (const float*)d_in[0];
  const float* biases  = (const float*)d_in[1];
  const float* factors = (const float*)d_in[2];
  float* out = (float*)d_out;
  float* ws  = (float*)d_ws;   // assumes ws_size >= 16*3*512*512*4 bytes

  dim3 grid(WW / TILE_X, HH / TILE_Y, BB * CC);   // 8 x 32 x 48
  dim3 block(256);

  const float* src = x;
  float* dst = ws;
  for (int it = 0; it < NUM_ITER; ++it) {
    AnisoDiff2D_pm_step<<<grid, block, 0, stream>>>(src, dst, factors, biases);
    src = dst;
    dst = (dst == ws) ? out : ws;
  }
  // NUM_ITER is even, so the final sweep wrote d_out.
}